// MaskedConvProjection_61040075210852
// MI455X (gfx1250) — compile-verified
//
#include <hip/hip_runtime.h>
#include <hip/hip_bf16.h>

// CDNA5 (gfx1250) causal-Toeplitz GEMM:
//   out[b,i] = sum_{j<=i} x[b,j] * kernel[i-j]   (f32 in, f32 out)
// = X[2048,4096] @ B[4096,4096], B[k,n] = kernel[n-k] (n>=k else 0).
//
// Compute-bound (~1000 FLOP/byte): math on V_WMMA_F32_16X16X4_F32 (exact f32).
// B is synthesized from a REVERSED LDS-resident kernel vector rk[i] =
// kernel[4095-i] with a 128-float zeroed tail guard: the two taps of each B
// fragment (kernel[d], kernel[d-1]) sit at consecutive ASCENDING LDS
// addresses, so each fragment is one ds_load_2addr_b32 whose 8-bit immediate
// offset fields encode the fragment position -- the per-chunk base index is
// laundered through an empty asm barrier so the compiler cannot fold the
// large compile-time bias into per-access displacements (which would exceed
// the 8-bit ds offset fields and spawn a v_add per fragment).
// Triangular structure -> k < n0+128 per block (2x work saving).

typedef __attribute__((ext_vector_type(2))) float v2f;
typedef __attribute__((ext_vector_type(8))) float v8f;

#define FEATS 4096      // N and K
#define NBATCH 2048     // M
#define BM 128
#define BN 128
#define BK 32
#define ASTRIDE 36      // 32 + pad: conflict-free frag reads, 16B-aligned fills
#define GUARD 128       // zeroed floats AFTER reversed kernel (absorbs d < 0)
#define RKN (FEATS + GUARD)
#define ATILE_F (BM * ASTRIDE)
#define BIAS2 112       // keeps all 2addr immediate offsets in [0,141]

__global__ __launch_bounds__(256)
void toeplitz_wmma_kernel(const float* __restrict__ x,
                          const float* __restrict__ kern,
                          float* __restrict__ out) {
    // single allocation: reversed kernel at LDS offset 0, X tiles after it
    __shared__ float smem[RKN + 2 * ATILE_F];        // 52.5 KB
    float* rk    = smem;
    float* abase = smem + RKN;                       // two BM x BK tile buffers

    const int tid  = threadIdx.x;
    const int lane = tid & 31;
    const int wv   = tid >> 5;          // wave 0..7: owns M rows [16*wv, 16*wv+16)
    const int ln   = lane & 15;         // N position / M row within fragment
    const int half = lane >> 4;         // 0: K pair {0,1}, 1: K pair {2,3}
    const int koff = half * 2;

    const int n0 = blockIdx.x * BN;     // output column tile origin
    const int m0 = blockIdx.y * BM;     // output row tile origin

    // ---- preload kernel REVERSED into LDS, zero the tail guard ----
    #pragma unroll
    for (int i = tid; i < FEATS / 4; i += 256) {
        float4 v = *(const float4*)(kern + i * 4);
        float4 r; r.x = v.w; r.y = v.z; r.z = v.y; r.w = v.x;
        *(float4*)&rk[FEATS - 4 - i * 4] = r;        // rk[4095-j] = kern[j]
    }
    if (tid < GUARD) rk[FEATS + tid] = 0.0f;

    // ---- stage first X tile: rows [m0,m0+128), cols [0,32) ----
    const int lr = tid >> 3;            // 0..31
    const int lq = tid & 7;             // float4 slot within a 32-float row
    #pragma unroll
    for (int p = 0; p < 4; ++p) {
        const int row = lr + p * 32;
        float4 v = *(const float4*)(x + (size_t)(m0 + row) * FEATS + lq * 4);
        *(float4*)&abase[row * ASTRIDE + lq * 4] = v;
    }
    __syncthreads();

    v8f acc[8];
    #pragma unroll
    for (int t = 0; t < 8; ++t) acc[t] = (v8f){0.f,0.f,0.f,0.f,0.f,0.f,0.f,0.f};

    // causality: columns n < n0+128 only need k <= n  ->  k < n0+128
    const int nchunks = (n0 + BN) / BK;

    // reversed-kernel base: rk index of kernel[d] is 4095-d;
    // d = n0+ln-koff-kc + (16t-4s)  =>  idx = rl0 + kc + (112+4s-16t), BIAS2 folded
    const int rl0   = (FEATS - 1 - BIAS2) + koff - ln - n0;
    const int mbase = (wv * 16 + ln) * ASTRIDE + koff;

    for (int c = 0; c < nchunks; ++c) {
        const int kc = c * BK;

        // ---- stage next X tile into the other buffer (+ speculative prefetch) ----
        if (c + 1 < nchunks) {
            const int kn = kc + BK;
            float* dst = abase + ((c + 1) & 1) * ATILE_F;
            #pragma unroll
            for (int p = 0; p < 4; ++p) {
                const int row = lr + p * 32;
                const float* src = x + (size_t)(m0 + row) * FEATS + kn + lq * 4;
                float4 v = *(const float4*)src;
                __builtin_prefetch(src + BK, 0, 1);   // gfx1250 global_prefetch_b8
                *(float4*)&dst[row * ASTRIDE + lq * 4] = v;
            }
        }

        // ---- compute: 8 K-steps x 8 N-subtiles of v_wmma_f32_16x16x4_f32 ----
        const float* abuf = abase + (c & 1) * ATILE_F;
        int rbase = rl0 + kc;
        // opaque base: forces the 3983-element bias into the base VGPR so all
        // per-fragment displacements (<= 141 elements) fit the ds 2addr fields
        asm volatile("" : "+v"(rbase));
        const float* rp = rk + rbase;
        #pragma unroll
        for (int s = 0; s < 8; ++s) {
            // A fragment (16x4 f32): v0 = A[m, kbase], v1 = A[m, kbase+1]
            float2 av = *(const float2*)&abuf[mbase + s * 4];
            v2f a; a.x = av.x; a.y = av.y;
            #pragma unroll
            for (int t = 0; t < 8; ++t) {
                // B[k,n] = kernel[n-k]; reversed layout -> ascending pair
                const int off = BIAS2 + 4 * s - 16 * t;   // compile-time, [0,140]
                v2f b; b.x = rp[off]; b.y = rp[off + 1];
                acc[t] = __builtin_amdgcn_wmma_f32_16x16x4_f32(
                    false, a, false, b, (short)0, acc[t], false, false);
            }
        }
        __syncthreads();
    }

    // ---- writeback: C/D layout (8 VGPRs, lanes 0-15 -> M=r, 16-31 -> M=r+8) ----
    #pragma unroll
    for (int t = 0; t < 8; ++t) {
        const int col = n0 + 16 * t + ln;
        #pragma unroll
        for (int r = 0; r < 8; ++r) {
            const int row = m0 + wv * 16 + half * 8 + r;
            out[(size_t)row * FEATS + col] = acc[t][r];
        }
    }
}

extern "C" void kernel_launch(void* const* d_in, const int* in_sizes, int n_in,
                              void* d_out, int out_size, void* d_ws, size_t ws_size,
                              hipStream_t stream) {
    const float* x    = (const float*)d_in[0];   // [2048, 4096] f32
    const float* kern = (const float*)d_in[1];   // [4096] f32
    float* out        = (float*)d_out;           // [2048, 4096] f32

    dim3 grid(FEATS / BN, NBATCH / BM);          // 32 x 16 workgroups
    toeplitz_wmma_kernel<<<grid, 256, 0, stream>>>(x, kern, out);
}